// SFA_Encoder_12841952215137
// MI455X (gfx1250) — compile-verified
//
#include <hip/hip_runtime.h>
#include <hip/hip_bf16.h>

// ---------------------------------------------------------------------------
// LightGCN-style propagation: acc = (I + A + A^2 + A^3) * ego / 4
// Memory-bound COO SpMM; edge stream staged into LDS via the CDNA5 Tensor
// Data Mover (TENSORcnt async DMA), gathers as float4, scatter via f32 atomics.
// NOTE: lgcn_spmm is defined first so the disasm snippet shows the hot kernel.
// ---------------------------------------------------------------------------

#define TILE 512        // edges staged per TDM tile
#define NT   2          // tiles per block (double buffered)
#define CHUNK (TILE * NT)

typedef unsigned int u32;
typedef __attribute__((ext_vector_type(4))) unsigned int u32x4;
typedef __attribute__((ext_vector_type(8))) int          i32x8;
typedef __attribute__((ext_vector_type(4))) int          i32x4;

struct __align__(16) SMem {
  int   rows[2][TILE];
  int   cols[2][TILE];
  float vals[2][TILE];
};
// byte offsets of the arrays inside the (single) static LDS allocation,
// which the compiler places at LDS offset 0.
#define SM_ROWS_OFF 0u
#define SM_COLS_OFF (2u * TILE * 4u)
#define SM_VALS_OFF (4u * TILE * 4u)

// The TDM writes LDS "behind the compiler's back" (the LDS address travels in
// an integer descriptor, not a pointer), so LLVM would otherwise prove the
// shared object is never stored and fold its loads to undef. Laundering the
// pointer through an asm with a memory clobber makes the object escape and
// marks it may-written, forcing real ds_load instructions after each tile.
__device__ __forceinline__ void lds_publish(void* p) {
  asm volatile("" : "+v"(p) : : "memory");
}

// 1-D contiguous TDM load: copy `tile_elems` dwords from global to LDS.
// tensor_dim0 = remaining elements so tail tiles OOB-read as zeros.
__device__ __forceinline__ void tdm_load_b32_1d(u32 lds_byte_off, const void* gptr,
                                                u32 tile_elems, u32 rem_elems) {
  unsigned long long ga = (unsigned long long)gptr;
  u32x4 g0;
  g0[0] = 1u;                                              // count=1 (valid), user mode
  g0[1] = lds_byte_off;                                    // lds_addr (bytes)
  g0[2] = (u32)(ga & 0xFFFFFFFFull);                       // global_addr[31:0]
  g0[3] = (u32)((ga >> 32) & 0x1FFFFFFull) | (2u << 30);   // global_addr[56:32] | type=2
  i32x8 g1;
  g1[0] = (int)(2u << 16);                                 // workgroup_mask=0, data_size=4B
  g1[1] = (int)((rem_elems & 0xFFFFu) << 16);              // tensor_dim0[15:0] at bits 63:48
  g1[2] = (int)(rem_elems >> 16);                          // tensor_dim0[31:16]; tensor_dim1=0
  g1[3] = (int)((tile_elems & 0xFFFFu) << 16);             // tile_dim0
  g1[4] = 0;                                               // tile_dim1 = tile_dim2 = 0 (1-D)
  g1[5] = (int)rem_elems;                                  // tensor_dim0_stride (benign, 1-D)
  g1[6] = 0;
  g1[7] = 0;
  i32x4 z4 = {0, 0, 0, 0};
#if __clang_major__ >= 23
  i32x8 z8 = {0, 0, 0, 0, 0, 0, 0, 0};
  __builtin_amdgcn_tensor_load_to_lds(g0, g1, z4, z4, z8, 0);
#else
  __builtin_amdgcn_tensor_load_to_lds(g0, g1, z4, z4, 0);
#endif
}

// ---------------------------------------------------------------------------
// COO SpMM: xout[r,:] += v * xin[c,:] for each edge (r,c,v).
// Edge triples streamed to LDS by the TDM (double buffered); 16 lanes per
// edge, one float4 (16B) per lane => one 256B coalesced gather per edge.
__global__ __launch_bounds__(256) void lgcn_spmm(const int* __restrict__ erow,
                                                 const int* __restrict__ ecol,
                                                 const float* __restrict__ eval,
                                                 const float* __restrict__ xin,
                                                 float* __restrict__ xout, int E) {
  __shared__ SMem sm;
  const int tid = threadIdx.x;
  const int lane16 = tid & 15;   // which float4 of the 64-float row
  const int grp = tid >> 4;      // edge slot group (0..15)
  const int chunk0 = (int)blockIdx.x * CHUNK;

  int ntiles = (E - chunk0 + TILE - 1) / TILE;
  if (ntiles > NT) ntiles = NT;
  if (ntiles <= 0) return;

  const bool issuer = (tid < 32);  // wave 0 drives the TDM (EXEC ignored by TDM)

  if (issuer) {
    const u32 rem = (u32)(E - chunk0);
    tdm_load_b32_1d(SM_ROWS_OFF, erow + chunk0, TILE, rem);
    tdm_load_b32_1d(SM_COLS_OFF, ecol + chunk0, TILE, rem);
    tdm_load_b32_1d(SM_VALS_OFF, eval + chunk0, TILE, rem);
  }

  for (int k = 0; k < ntiles; ++k) {
    const int buf = k & 1;
    if (issuer) {
      if (k + 1 < ntiles) {
        const int tb = chunk0 + (k + 1) * TILE;
        const u32 rem = (u32)(E - tb);
        const u32 off = (u32)((k + 1) & 1) * TILE * 4u;
        tdm_load_b32_1d(SM_ROWS_OFF + off, erow + tb, TILE, rem);
        tdm_load_b32_1d(SM_COLS_OFF + off, ecol + tb, TILE, rem);
        tdm_load_b32_1d(SM_VALS_OFF + off, eval + tb, TILE, rem);
        __builtin_amdgcn_s_wait_tensorcnt(3);  // oldest tile (3 DMAs) complete
      } else {
        __builtin_amdgcn_s_wait_tensorcnt(0);
      }
    }
    __syncthreads();   // publish LDS tile to all 8 waves
    lds_publish(&sm);  // compiler-visible "the TDM wrote this" barrier

    const int base = chunk0 + k * TILE;
#pragma unroll 4
    for (int e = grp; e < TILE; e += 16) {
      const int eg = base + e;
      if (eg < E) {
        const int r = sm.rows[buf][e];   // ds_load_b32
        const int c = sm.cols[buf][e];
        const float v = sm.vals[buf][e];
        const float4 g =
            ((const float4*)(xin + (size_t)c * 64))[lane16];  // global_load_b128
        float* dst = xout + (size_t)r * 64 + (size_t)lane16 * 4;
        atomicAdd(dst + 0, v * g.x);  // non-returning global_atomic_add_f32
        atomicAdd(dst + 1, v * g.y);
        atomicAdd(dst + 2, v * g.z);
        atomicAdd(dst + 3, v * g.w);
      }
    }
    __syncthreads();  // all reads done before TDM may overwrite this buffer
  }
}

// ---------------------------------------------------------------------------
// init: ego0 = concat(user,item); acc(d_out) = same
__global__ __launch_bounds__(256) void lgcn_init(const float4* __restrict__ ue,
                                                 const float4* __restrict__ ie,
                                                 float4* __restrict__ ego,
                                                 float4* __restrict__ acc,
                                                 int n4_user, int n4_total) {
  for (int i = blockIdx.x * blockDim.x + threadIdx.x; i < n4_total;
       i += gridDim.x * blockDim.x) {
    float4 v = (i < n4_user) ? ue[i] : ie[i - n4_user];
    ego[i] = v;
    acc[i] = v;
  }
}

__global__ __launch_bounds__(256) void lgcn_zero(float4* __restrict__ p, int n4) {
  float4 z = make_float4(0.f, 0.f, 0.f, 0.f);
  for (int i = blockIdx.x * blockDim.x + threadIdx.x; i < n4;
       i += gridDim.x * blockDim.x)
    p[i] = z;
}

// acc = (acc + ego_new) * scale ; clear the buffer that becomes the next
// layer's atomic-scatter target (folds the per-layer zeroing pass away).
__global__ __launch_bounds__(256) void lgcn_acc_clear(float4* __restrict__ acc,
                                                      const float4* __restrict__ ego_new,
                                                      float4* __restrict__ clearbuf,
                                                      int n4, float scale) {
  float4 z = make_float4(0.f, 0.f, 0.f, 0.f);
  for (int i = blockIdx.x * blockDim.x + threadIdx.x; i < n4;
       i += gridDim.x * blockDim.x) {
    float4 a = acc[i];
    float4 e = ego_new[i];
    a.x = (a.x + e.x) * scale;
    a.y = (a.y + e.y) * scale;
    a.z = (a.z + e.z) * scale;
    a.w = (a.w + e.w) * scale;
    acc[i] = a;
    clearbuf[i] = z;
  }
}

// ---------------------------------------------------------------------------
extern "C" void kernel_launch(void* const* d_in, const int* in_sizes, int n_in,
                              void* d_out, int out_size, void* d_ws, size_t ws_size,
                              hipStream_t stream) {
  const float* user_emb = (const float*)d_in[0];
  const float* item_emb = (const float*)d_in[1];
  const int*   erow     = (const int*)d_in[2];
  const int*   ecol     = (const int*)d_in[3];
  const float* eval     = (const float*)d_in[4];
  float*       out      = (float*)d_out;

  const int UD = in_sizes[0];   // U*D = 3,200,000 floats
  const int ND = out_size;      // N*D = 9,600,000 floats
  const int E  = in_sizes[2];   // 4,000,000 edges

  float* ego0 = (float*)d_ws;          // 38.4 MB
  float* ego1 = ego0 + (size_t)ND;     // 38.4 MB

  const int n4 = ND / 4;
  dim3 blk(256);
  const int gElem = (n4 + 255) / 256;
  const int gSp   = (E + CHUNK - 1) / CHUNK;

  lgcn_init<<<gElem, blk, 0, stream>>>((const float4*)user_emb, (const float4*)item_emb,
                                       (float4*)ego0, (float4*)out, UD / 4, n4);
  lgcn_zero<<<gElem, blk, 0, stream>>>((float4*)ego1, n4);

  float* cur = ego0;
  float* nxt = ego1;
  for (int l = 0; l < 3; ++l) {
    lgcn_spmm<<<gSp, blk, 0, stream>>>(erow, ecol, eval, cur, nxt, E);
    const float scale = (l == 2) ? 0.25f : 1.0f;
    lgcn_acc_clear<<<gElem, blk, 0, stream>>>((float4*)out, (const float4*)nxt,
                                              (float4*)cur, n4, scale);
    float* t = cur; cur = nxt; nxt = t;  // ping-pong (deterministic host-side swap)
  }
}